// Effcient_GPsampler_19696720019897
// MI455X (gfx1250) — compile-verified
//
#include <hip/hip_runtime.h>
#include <hip/hip_bf16.h>

typedef float v8f __attribute__((ext_vector_type(8)));
typedef float v2f __attribute__((ext_vector_type(2)));
typedef unsigned int u32x4 __attribute__((ext_vector_type(4)));
typedef int i32x4 __attribute__((ext_vector_type(4)));
typedef int i32x8 __attribute__((ext_vector_type(8)));

#ifndef __has_builtin
#define __has_builtin(x) 0
#endif
#if defined(__gfx1250__) && __has_builtin(__builtin_amdgcn_tensor_load_to_lds) && \
    __has_builtin(__builtin_amdgcn_s_wait_tensorcnt)
#define USE_TDM 1
#else
#define USE_TDM 0
#endif

// ---------------------------------------------------------------------------
// Stage 1: density + datarepr via RBF kernel, C-reduction on the matrix unit.
//   density[b,g,H]  = sum_c exp(coef_H * (xg[b,g,H]-xc[b,c,H])^2)
//   repr[b,g,H]     = (sum_c K*yc[b,c,H]) / (density + 1e-4)
// One block = one (b,H) pair slab; 8 waves, each owns a 16-row g tile.
// WMMA f32 16x16x4: A = K tile (16g x 4c), B col0 = ones, col1 = yc chunk.
// xc/yc rows staged into LDS by the Tensor Data Mover (stride-8 column
// gather as a 1x512 TDM tile), falling back to per-thread loads.
// ---------------------------------------------------------------------------

#if USE_TDM
__device__ __forceinline__ void tdm_gather_column(const float* gsrc,
                                                  unsigned int lds_byte_off) {
    // D# group0: count=1 | lds_addr | global_addr[56:0] | type=2 ("image")
    const unsigned long long ga = (unsigned long long)(const void*)gsrc;
    u32x4 g0;
    g0.x = 1u;                                        // count=1, user descriptor
    g0.y = lds_byte_off;                              // LDS byte address
    g0.z = (unsigned int)(ga & 0xffffffffull);        // global_addr[31:0]
    g0.w = (unsigned int)((ga >> 32) & 0x01ffffffull) // global_addr[56:32]
           | (2u << 30);                              // type=2
    // D# group1: data_size=4B; tensor 1 x 512 (dim0 x dim1), dim0_stride=8
    // elements; tile 1 x 512 -> 512 contiguous floats in LDS.
    i32x8 g1;
    g1[0] = (int)(2u << 16);      // wg_mask=0, data_size=2 (4 bytes)
    g1[1] = (int)(1u << 16);      // tensor_dim0[15:0] = 1
    g1[2] = (int)(512u << 16);    // tensor_dim0 hi = 0, tensor_dim1[15:0] = 512
    g1[3] = (int)(1u << 16);      // tensor_dim1 hi = 0, tile_dim0 = 1
    g1[4] = 512;                  // tile_dim1 = 512, tile_dim2 = 0
    g1[5] = 8;                    // tensor_dim0_stride = 8 (elements)
    g1[6] = 0;                    // stride hi, tensor_dim1_stride lo
    g1[7] = 0;
    i32x4 z4 = {0, 0, 0, 0};      // groups 2/3: dims 2..4 unused (zero)
#if __clang_major__ >= 23
    i32x8 z8 = {0, 0, 0, 0, 0, 0, 0, 0};
    __builtin_amdgcn_tensor_load_to_lds(g0, g1, z4, z4, z8, 0);
#else
    __builtin_amdgcn_tensor_load_to_lds(g0, g1, z4, z4, 0);
#endif
}
#endif

__global__ __launch_bounds__(256) void gp_density_repr(
    const float* __restrict__ xc, const float* __restrict__ yc,
    const float* __restrict__ xgrid, const float* __restrict__ log_ls,
    float* __restrict__ out_density, float* __restrict__ out_repr)
{
    __shared__ float s_xc[512];
    __shared__ float s_yc[512];

    const int tid = threadIdx.x;
    const int p = blockIdx.y;          // 0..63
    const int b = p >> 3;
    const int H = p & 7;

#if USE_TDM
    if (tid < 32) {                    // wave 0 drives the DMA engine
        tdm_gather_column(xc + ((size_t)b * 512) * 8 + H,
                          (unsigned int)(size_t)(void*)s_xc);
        tdm_gather_column(yc + ((size_t)b * 512) * 8 + H,
                          (unsigned int)(size_t)(void*)s_yc);
        __builtin_amdgcn_s_wait_tensorcnt(0);
    }
#else
    for (int i = tid; i < 512; i += 256) {
        s_xc[i] = xc[(b * 512 + i) * 8 + H];
        s_yc[i] = yc[(b * 512 + i) * 8 + H];
    }
#endif
    const float coef = -0.5f * __expf(-2.0f * log_ls[H]);   // -0.5/ls^2
    __syncthreads();

    const int lane = tid & 31;
    const int wave = tid >> 5;
    const int m  = lane & 15;          // A row / B,C,D column index
    const int hi = lane >> 4;          // half-wave selector
    const int g0 = (blockIdx.x * 8 + wave) * 16;

    // A-row coordinate for this lane (xgrid shape (8,2048,1,8))
    const float xg = xgrid[(b * 2048 + g0 + m) * 8 + H];

    // Branchless B selectors: column n==0 -> ones, n==1 -> yc, else 0.
    const float sel0 = (m == 0) ? 1.0f : 0.0f;
    const float sel1 = (m == 1) ? 1.0f : 0.0f;

    v8f acc = {0.f, 0.f, 0.f, 0.f, 0.f, 0.f, 0.f, 0.f};

#pragma unroll 4
    for (int c0 = 0; c0 < 512; c0 += 4) {
        const int cb = c0 + 2 * hi;    // ISA A layout: lane half holds K = 2*hi, 2*hi+1
        const float x0 = s_xc[cb];
        const float x1 = s_xc[cb + 1];
        const float y0 = s_yc[cb];
        const float y1 = s_yc[cb + 1];
        const float dx0 = xg - x0;
        const float dx1 = xg - x1;
        v2f A, B;
        A.x = __expf(coef * dx0 * dx0);
        A.y = __expf(coef * dx1 * dx1);
        B.x = fmaf(sel1, y0, sel0);
        B.y = fmaf(sel1, y1, sel0);
        acc = __builtin_amdgcn_wmma_f32_16x16x4_f32(
            false, A, false, B, (short)0, acc, false, false);
    }

    // C/D layout: VGPR v holds row M = v + 8*hi, column N = lane&15.
    // Column 0 = density, column 1 = numerator.
#pragma unroll
    for (int v = 0; v < 8; ++v) {
        const float cv   = acc[v];
        const float dens = __shfl(cv, (hi << 4) | 0, 32);
        const float num  = __shfl(cv, (hi << 4) | 1, 32);
        if (m == 0) {
            const int g = g0 + 8 * hi + v;
            const int o = (b * 2048 + g) * 8 + H;
            out_density[o] = dens;
            out_repr[o]    = num / (dens + 1e-4f);
        }
    }
}

// ---------------------------------------------------------------------------
// Stage 2: tiny conv net + softmax/gumbel tail. One block per batch element.
// Ping-pongs activations between two LDS buffers (61 KB total).
// ---------------------------------------------------------------------------
__global__ __launch_bounds__(256) void mixnet(
    const float* __restrict__ repr,   // datarepr (8,2048,8); x[b,i,g]=repr[(b*2048+g)*8+i]
    const float* __restrict__ w1, const float* __restrict__ b1,
    const float* __restrict__ w2, const float* __restrict__ b2,
    const float* __restrict__ w3, const float* __restrict__ b3,
    const float* __restrict__ w4, const float* __restrict__ b4,
    const float* __restrict__ wl, const float* __restrict__ bl,
    const float* __restrict__ unif,
    float* __restrict__ out_mixw, float* __restrict__ out_samp)
{
    __shared__ float s_a[10240];      // h1 (10x1022), later h3 (10x253)
    __shared__ float s_b[5120];       // h2 (10x509),  later h4 (10x125)
    __shared__ float s_h[10];
    __shared__ float s_tl[32];

    const int tid = threadIdx.x;
    const int b = blockIdx.x;

    // conv1: (8,2048) -> relu -> (10,1022), stride 2, K=5
    for (int idx = tid; idx < 10 * 1022; idx += 256) {
        const int o = idx / 1022, l = idx % 1022;
        float s = b1[o];
        for (int i = 0; i < 8; ++i)
            for (int k = 0; k < 5; ++k)
                s += repr[(b * 2048 + 2 * l + k) * 8 + i] * w1[(o * 8 + i) * 5 + k];
        s_a[idx] = fmaxf(s, 0.0f);
    }
    __syncthreads();

    // conv2: (10,1022) -> relu -> (10,509)
    for (int idx = tid; idx < 10 * 509; idx += 256) {
        const int o = idx / 509, l = idx % 509;
        float s = b2[o];
        for (int i = 0; i < 10; ++i)
            for (int k = 0; k < 5; ++k)
                s += s_a[i * 1022 + 2 * l + k] * w2[(o * 10 + i) * 5 + k];
        s_b[idx] = fmaxf(s, 0.0f);
    }
    __syncthreads();

    // conv3: (10,509) -> relu -> (10,253), overwrite s_a
    for (int idx = tid; idx < 10 * 253; idx += 256) {
        const int o = idx / 253, l = idx % 253;
        float s = b3[o];
        for (int i = 0; i < 10; ++i)
            for (int k = 0; k < 5; ++k)
                s += s_b[i * 509 + 2 * l + k] * w3[(o * 10 + i) * 5 + k];
        s_a[idx] = fmaxf(s, 0.0f);
    }
    __syncthreads();

    // conv4: (10,253) -> (10,125), no relu, overwrite s_b
    for (int idx = tid; idx < 10 * 125; idx += 256) {
        const int o = idx / 125, l = idx % 125;
        float s = b4[o];
        for (int i = 0; i < 10; ++i)
            for (int k = 0; k < 5; ++k)
                s += s_a[i * 253 + 2 * l + k] * w4[(o * 10 + i) * 5 + k];
        s_b[idx] = s;
    }
    __syncthreads();

    // adaptive avg pool (mean over 125)
    if (tid < 10) {
        float s = 0.f;
        for (int l = 0; l < 125; ++l) s += s_b[tid * 125 + l];
        s_h[tid] = s * (1.0f / 125.0f);
    }
    __syncthreads();

    // linear: (10) -> (32) raw logits
    if (tid < 32) {
        float s = bl[tid];
        for (int j = 0; j < 10; ++j) s += s_h[j] * wl[tid * 10 + j];
        s_tl[tid] = s;
    }
    __syncthreads();

    // tempering + softmax per channel group of nmix=4; keep tempered logits
    if (tid < 8) {
        const int H = tid;
        float mx = s_tl[H * 4];
        for (int k = 1; k < 4; ++k) mx = fmaxf(mx, s_tl[H * 4 + k]);
        float t[4], e[4], ssum = 0.f;
        for (int k = 0; k < 4; ++k) t[k] = (s_tl[H * 4 + k] - mx) * 10.0f; // /0.1
        for (int k = 0; k < 4; ++k) { e[k] = __expf(t[k]); ssum += e[k]; }
        for (int k = 0; k < 4; ++k) out_mixw[(b * 8 + H) * 4 + k] = e[k] / ssum;
        for (int k = 0; k < 4; ++k) s_tl[H * 4 + k] = t[k];  // tempered, for gumbel
    }
    __syncthreads();

    // gumbel softmax samples: (ns=10, ch=8, nmix=4)
    for (int idx = tid; idx < 80; idx += 256) {
        const int s = idx / 8, H = idx % 8;
        float hh[4], mx = -1e30f;
        for (int k = 0; k < 4; ++k) {
            const float u = unif[((b * 10 + s) * 8 + H) * 4 + k];
            const float g = -__logf(-__logf(u + 1e-8f));
            hh[k] = (g + s_tl[H * 4 + k]) * 10.0f;   // / GUMBEL_TEMP
            mx = fmaxf(mx, hh[k]);
        }
        float e[4], ssum = 0.f;
        for (int k = 0; k < 4; ++k) { e[k] = __expf(hh[k] - mx); ssum += e[k]; }
        for (int k = 0; k < 4; ++k) {
            float y = e[k] / ssum;
            y = fminf(fmaxf(y, 1e-8f), 1.0f - 1e-8f);
            out_samp[((b * 10 + s) * 8 + H) * 4 + k] = y;
        }
    }
}

extern "C" void kernel_launch(void* const* d_in, const int* in_sizes, int n_in,
                              void* d_out, int out_size, void* d_ws, size_t ws_size,
                              hipStream_t stream) {
    const float* xc     = (const float*)d_in[0];
    const float* yc     = (const float*)d_in[1];
    const float* xgrid  = (const float*)d_in[2];
    const float* log_ls = (const float*)d_in[3];
    const float* w1 = (const float*)d_in[4];  const float* b1 = (const float*)d_in[5];
    const float* w2 = (const float*)d_in[6];  const float* b2 = (const float*)d_in[7];
    const float* w3 = (const float*)d_in[8];  const float* b3 = (const float*)d_in[9];
    const float* w4 = (const float*)d_in[10]; const float* b4 = (const float*)d_in[11];
    const float* wl = (const float*)d_in[12]; const float* bl = (const float*)d_in[13];
    const float* unif = (const float*)d_in[14];

    float* out = (float*)d_out;
    float* out_density = out;              // (8,2048,8)  = 131072
    float* out_repr    = out + 131072;     // (8,2048,8)  = 131072
    float* out_mixw    = out + 262144;     // (8,8,4)     = 256
    float* out_samp    = out + 262400;     // (8,10,8,4)  = 2560

    dim3 g1(16, 64);                        // 16 g-slabs x 64 (b,H) pairs
    gp_density_repr<<<g1, dim3(256), 0, stream>>>(
        xc, yc, xgrid, log_ls, out_density, out_repr);

    mixnet<<<dim3(8), dim3(256), 0, stream>>>(
        out_repr, w1, b1, w2, b2, w3, b3, w4, b4, wl, bl, unif,
        out_mixw, out_samp);
}